// AtomicGeometricInvariantHead_60189671686869
// MI455X (gfx1250) — compile-verified
//
#include <hip/hip_runtime.h>

#define NPTS 32768
#define KNN 16
#define KMAXS 17
#define BATCH 8
#define FEPS 1e-6f

typedef __attribute__((ext_vector_type(2))) float v2f;
typedef __attribute__((ext_vector_type(8))) float v8f;

// Workspace layout: per batch, 36 doubles:
// [0..2]  sumX, sumY, sumZ
// [3..8]  sum xx, xy, xz, yy, yz, zz
// [9]     sum d (over N*K)
// [10..19] sums: tr, tr^2, anis, anis^2, vol, vol^2, trd, trd^2, anisd, anisd^2
// [20..35] rbf bin sums (rbf*w summed over N*K)
#define WS_PER_B 36

__global__ void zero_ws_kernel(double* ws) {
  int i = blockIdx.x * blockDim.x + threadIdx.x;
  if (i < BATCH * WS_PER_B) ws[i] = 0.0;
}

// ---------------- Pass 1: global-cov moment sums + sum of neighbor distances --
__global__ __launch_bounds__(256) void pass1_kernel(const float* __restrict__ xyz,
                                                    const int* __restrict__ idx,
                                                    double* __restrict__ ws) {
  __shared__ float red[256];
  const int tid = threadIdx.x;
  const int b = blockIdx.x >> 7;                  // 128 blocks per batch
  const int p = ((blockIdx.x & 127) << 8) | tid;  // point index in [0, 32768)
  const float* X = xyz + (size_t)b * NPTS * 3;
  const int* I = idx + ((size_t)b * NPTS + (size_t)p) * KMAXS + 1;

  const float px = X[p * 3 + 0], py = X[p * 3 + 1], pz = X[p * 3 + 2];

  float acc[10];
  acc[0] = px; acc[1] = py; acc[2] = pz;
  acc[3] = px * px; acc[4] = px * py; acc[5] = px * pz;
  acc[6] = py * py; acc[7] = py * pz; acc[8] = pz * pz;

  float sd = 0.f;
  for (int k = 0; k < KNN; ++k) {
    const int j = I[k];
    const float rx = X[j * 3 + 0] - px;
    const float ry = X[j * 3 + 1] - py;
    const float rz = X[j * 3 + 2] - pz;
    const float d2 = rx * rx + ry * ry + rz * rz;
    sd += sqrtf(d2 + FEPS);
  }
  acc[9] = sd;

  double* out = ws + b * WS_PER_B;
  for (int q = 0; q < 10; ++q) {
    red[tid] = acc[q];
    __syncthreads();
    for (int s = 128; s > 0; s >>= 1) {
      if (tid < s) red[tid] += red[tid + s];
      __syncthreads();
    }
    if (tid == 0) atomicAdd(&out[q], (double)red[0]);
    __syncthreads();
  }
}

// ---------------- Pass 2: per-point weighted covariance invariants + RBF ------
__global__ __launch_bounds__(256) void pass2_kernel(const float* __restrict__ xyz,
                                                    const int* __restrict__ idx,
                                                    double* __restrict__ ws) {
  __shared__ float red[256];
  const int tid = threadIdx.x;
  const int b = blockIdx.x >> 7;
  const int p = ((blockIdx.x & 127) << 8) | tid;
  const float* X = xyz + (size_t)b * NPTS * 3;
  const int* I = idx + ((size_t)b * NPTS + (size_t)p) * KMAXS + 1;
  double* wsb = ws + b * WS_PER_B;

  const float dmean = fmaxf((float)(wsb[9] / ((double)NPTS * (double)KNN)), FEPS);
  const float sigma = 0.75f * dmean;
  const float winv = 1.0f / (2.0f * sigma * sigma + FEPS);
  const float dninv = 1.0f / dmean;

  const float px = X[p * 3 + 0], py = X[p * 3 + 1], pz = X[p * 3 + 2];

  float C[6] = {0, 0, 0, 0, 0, 0};
  float U[6] = {0, 0, 0, 0, 0, 0};
  float wsum = 0.f;
  float rbf[16];
#pragma unroll
  for (int j = 0; j < 16; ++j) rbf[j] = 0.f;

  for (int k = 0; k < KNN; ++k) {
    const int jn = I[k];
    const float rx = X[jn * 3 + 0] - px;
    const float ry = X[jn * 3 + 1] - py;
    const float rz = X[jn * 3 + 2] - pz;
    const float d2 = rx * rx + ry * ry + rz * rz;
    const float d = sqrtf(d2 + FEPS);
    const float w = expf(-d2 * winv);
    wsum += w;
    C[0] += w * rx * rx; C[1] += w * rx * ry; C[2] += w * rx * rz;
    C[3] += w * ry * ry; C[4] += w * ry * rz; C[5] += w * rz * rz;
    const float ui = 1.0f / (d + FEPS);
    const float ux = rx * ui, uy = ry * ui, uz = rz * ui;
    U[0] += w * ux * ux; U[1] += w * ux * uy; U[2] += w * ux * uz;
    U[3] += w * uy * uy; U[4] += w * uy * uz; U[5] += w * uz * uz;
    const float dn = d * dninv;
#pragma unroll
    for (int j = 0; j < 16; ++j) {
      const float t = (dn - (float)j * (2.0f / 15.0f)) * (1.0f / 0.15f);
      rbf[j] += expf(-0.5f * t * t) * w;
    }
  }

  const float sc = 1.0f / (wsum + FEPS);
  const float c00 = C[0] * sc, c01 = C[1] * sc, c02 = C[2] * sc;
  const float c11 = C[3] * sc, c12 = C[4] * sc, c22 = C[5] * sc;
  const float tr = c00 + c11 + c22;
  const float fro2 = c00 * c00 + c11 * c11 + c22 * c22 +
                     2.f * (c01 * c01 + c02 * c02 + c12 * c12);
  float det = c00 * (c11 * c22 - c12 * c12) - c01 * (c01 * c22 - c12 * c02) +
              c02 * (c01 * c12 - c11 * c02);
  det = fmaxf(det, 0.f);
  const float anis = fro2 / (tr * tr + FEPS);
  const float vol = det / (tr * tr * tr + FEPS);

  const float u00 = U[0] * sc, u01 = U[1] * sc, u02 = U[2] * sc;
  const float u11 = U[3] * sc, u12 = U[4] * sc, u22 = U[5] * sc;
  const float trd = u00 + u11 + u22;
  const float fro2d = u00 * u00 + u11 * u11 + u22 * u22 +
                      2.f * (u01 * u01 + u02 * u02 + u12 * u12);
  const float anisd = fro2d / (trd * trd + FEPS);

  float acc[26];
  acc[0] = tr;    acc[1] = tr * tr;
  acc[2] = anis;  acc[3] = anis * anis;
  acc[4] = vol;   acc[5] = vol * vol;
  acc[6] = trd;   acc[7] = trd * trd;
  acc[8] = anisd; acc[9] = anisd * anisd;
#pragma unroll
  for (int j = 0; j < 16; ++j) acc[10 + j] = rbf[j];

  for (int q = 0; q < 26; ++q) {
    red[tid] = acc[q];
    __syncthreads();
    for (int s = 128; s > 0; s >>= 1) {
      if (tid < s) red[tid] += red[tid + s];
      __syncthreads();
    }
    if (tid == 0) atomicAdd(&wsb[10 + q], (double)red[0]);
    __syncthreads();
  }
}

// ---------------- Pass 3: finalize features + WMMA MLP (single wave32) -------
__global__ __launch_bounds__(32) void finalize_kernel(
    const double* __restrict__ ws, const float* __restrict__ W1,
    const float* __restrict__ b1, const float* __restrict__ gamma,
    const float* __restrict__ beta, const float* __restrict__ W2,
    const float* __restrict__ b2, float* __restrict__ out) {
  __shared__ float rawS[16][32];
  __shared__ float hS[16][32];
  const int lane = threadIdx.x;

  for (int r = 0; r < 16; ++r) rawS[r][lane] = 0.f;
  __syncthreads();

  if (lane < BATCH) {
    const double* a = ws + lane * WS_PER_B;
    const double Nn = (double)NPTS;
    const double mx = a[0] / Nn, my = a[1] / Nn, mz = a[2] / Nn;
    const double dv = Nn + 1e-6;
    const double g00 = (a[3] - Nn * mx * mx) / dv;
    const double g01 = (a[4] - Nn * mx * my) / dv;
    const double g02 = (a[5] - Nn * mx * mz) / dv;
    const double g11 = (a[6] - Nn * my * my) / dv;
    const double g12 = (a[7] - Nn * my * mz) / dv;
    const double g22 = (a[8] - Nn * mz * mz) / dv;
    const double trg = g00 + g11 + g22;
    const double fro2g = g00 * g00 + g11 * g11 + g22 * g22 +
                         2.0 * (g01 * g01 + g02 * g02 + g12 * g12);
    double detg = g00 * (g11 * g22 - g12 * g12) - g01 * (g01 * g22 - g12 * g02) +
                  g02 * (g01 * g12 - g11 * g02);
    if (detg < 0.0) detg = 0.0;
    rawS[lane][0] = (float)trg;
    rawS[lane][1] = (float)(fro2g / (trg * trg + 1e-6));
    rawS[lane][2] = (float)(detg / (trg * trg * trg + 1e-6));

    const double* S = a + 10;
    for (int q = 0; q < 5; ++q) {
      const double s1 = S[2 * q], s2 = S[2 * q + 1];
      const double mean = s1 / Nn;
      double var = (s2 - s1 * s1 / Nn) / (Nn - 1.0);
      if (var < 0.0) var = 0.0;
      rawS[lane][3 + 2 * q] = (float)mean;
      rawS[lane][4 + 2 * q] = (float)sqrt(var);
    }
    for (int j = 0; j < 16; ++j)
      rawS[lane][13 + j] = (float)(a[20 + j] / (Nn * (double)KNN));
  }
  __syncthreads();

  const int nl = lane & 15;
  const int khalf = (lane < 16) ? 0 : 2;

  // Layer 1: h = raw @ W1^T + b1   (16x32 tile; rows 8..15 are zero padding)
  for (int ntile = 0; ntile < 2; ++ntile) {
    const int n = ntile * 16 + nl;
    v8f c;
    const float cb = b1[n];
#pragma unroll
    for (int v = 0; v < 8; ++v) c[v] = cb;
#pragma unroll
    for (int kk = 0; kk < 8; ++kk) {
      const int kb = kk * 4 + khalf;
      v2f A, Bv;
      A[0] = rawS[nl][kb];
      A[1] = rawS[nl][kb + 1];
      Bv[0] = (kb < 29) ? W1[n * 29 + kb] : 0.f;
      Bv[1] = (kb + 1 < 29) ? W1[n * 29 + kb + 1] : 0.f;
      c = __builtin_amdgcn_wmma_f32_16x16x4_f32(false, A, false, Bv, (short)0, c,
                                                false, false);
    }
    const int rb = (lane < 16) ? 0 : 8;
#pragma unroll
    for (int v = 0; v < 8; ++v) hS[rb + v][n] = c[v];
  }
  __syncthreads();

  // LayerNorm + SiLU per row (feature dim = 32)
  if (lane < 16) {
    float m = 0.f;
    for (int j = 0; j < 32; ++j) m += hS[lane][j];
    m *= (1.f / 32.f);
    float var = 0.f;
    for (int j = 0; j < 32; ++j) {
      const float t = hS[lane][j] - m;
      var += t * t;
    }
    var *= (1.f / 32.f);
    const float rs = rsqrtf(var + 1e-5f);
    for (int j = 0; j < 32; ++j) {
      const float hv = (hS[lane][j] - m) * rs * gamma[j] + beta[j];
      hS[lane][j] = hv / (1.f + expf(-hv));  // SiLU
    }
  }
  __syncthreads();

  // Layer 2: out = h @ W2^T + b2
  for (int ntile = 0; ntile < 2; ++ntile) {
    const int n = ntile * 16 + nl;
    v8f c;
    const float cb = b2[n];
#pragma unroll
    for (int v = 0; v < 8; ++v) c[v] = cb;
#pragma unroll
    for (int kk = 0; kk < 8; ++kk) {
      const int kb = kk * 4 + khalf;
      v2f A, Bv;
      A[0] = hS[nl][kb];
      A[1] = hS[nl][kb + 1];
      Bv[0] = W2[n * 32 + kb];
      Bv[1] = W2[n * 32 + kb + 1];
      c = __builtin_amdgcn_wmma_f32_16x16x4_f32(false, A, false, Bv, (short)0, c,
                                                false, false);
    }
    if (lane < 16) {  // lanes 0..15 hold rows M=0..7 in c[0..7]
#pragma unroll
      for (int v = 0; v < 8; ++v) out[v * 32 + n] = c[v];
    }
  }
}

extern "C" void kernel_launch(void* const* d_in, const int* in_sizes, int n_in,
                              void* d_out, int out_size, void* d_ws, size_t ws_size,
                              hipStream_t stream) {
  (void)in_sizes; (void)n_in; (void)out_size; (void)ws_size;
  const float* xyz = (const float*)d_in[0];
  const int* idx = (const int*)d_in[1];
  const float* W1 = (const float*)d_in[2];
  const float* b1 = (const float*)d_in[3];
  const float* gamma = (const float*)d_in[4];
  const float* beta = (const float*)d_in[5];
  const float* W2 = (const float*)d_in[6];
  const float* b2 = (const float*)d_in[7];
  double* ws = (double*)d_ws;
  float* out = (float*)d_out;

  zero_ws_kernel<<<1, 512, 0, stream>>>(ws);
  pass1_kernel<<<BATCH * 128, 256, 0, stream>>>(xyz, idx, ws);
  pass2_kernel<<<BATCH * 128, 256, 0, stream>>>(xyz, idx, ws);
  finalize_kernel<<<1, 32, 0, stream>>>(ws, W1, b1, gamma, beta, W2, b2, out);
}